// GHMDice_55929064129141
// MI455X (gfx1250) — compile-verified
//
#include <hip/hip_runtime.h>

// ---------------------------------------------------------------------------
// GHM-Dice loss for MI455X (gfx1250, wave32).
// Two streaming passes (memory-bound: 805 MB @ 23.3 TB/s ~= 35 us floor):
//   pass1: I = sum(p*t), sum(p), sum(t), tot = sum(lw>0)      -> scale = 2I/S
//   pass2: per-bin counts (uint, exact) and per-bin sum(p*t)
//   finalize: loss = 1 - [2*(tot/maxn) * sum_b ptsum[b]/max(cnt[b],1)] / S
// Wave-level reductions use V_WMMA_F32_16X16X4_F32 with an all-ones B matrix:
// D[m][n] = v[m] + v[m+16]  (exact fp32, RNE) -> sum 8 acc VGPRs -> shfl_xor(16).
// ---------------------------------------------------------------------------

typedef __attribute__((ext_vector_type(2))) float v2f;
typedef __attribute__((ext_vector_type(8))) float v8f;

__device__ __forceinline__ float wave_sum_f32(float v) {
#if __has_builtin(__builtin_amdgcn_wmma_f32_16x16x4_f32)
  // A (16x4 f32): lane m in 0..15 supplies A[m][0]=v, A[m][1]=0;
  //               lane m+16 supplies A[m][2]=v, A[m][3]=0.
  // B = all ones (4x16) -> D[m][n] = v[lane m] + v[lane m+16].
  v2f a; a.x = v;    a.y = 0.0f;
  v2f b; b.x = 1.0f; b.y = 1.0f;
  v8f c = {};
  c = __builtin_amdgcn_wmma_f32_16x16x4_f32(
      /*neg_a=*/false, a, /*neg_b=*/false, b,
      /*c_mod=*/(short)0, c, /*reuse_a=*/false, /*reuse_b=*/false);
  // Lane n (n<16): sum_{M=0..7} rowsum_M ; lane n+16: sum_{M=8..15} rowsum_M.
  float s = c[0] + c[1] + c[2] + c[3] + c[4] + c[5] + c[6] + c[7];
  s += __shfl_xor(s, 16, 32);
  return s;  // full 32-lane sum, broadcast to all lanes
#else
  for (int off = 16; off > 0; off >>= 1) v += __shfl_xor(v, off, 32);
  return v;
#endif
}

__device__ __forceinline__ unsigned int wave_sum_u32(unsigned int v) {
  for (int off = 16; off > 0; off >>= 1) v += __shfl_xor(v, off, 32);
  return v;
}

// ------------------------------- pass 1 ------------------------------------
__global__ __launch_bounds__(256) void ghm_pass1(
    const float* __restrict__ P, const float* __restrict__ T,
    const float* __restrict__ W, long long N, float4* __restrict__ part) {
  const long long tid    = (long long)blockIdx.x * 256 + threadIdx.x;
  const long long stride = (long long)gridDim.x * 256;
  const long long N4     = N >> 2;
  const float4* P4 = (const float4*)P;
  const float4* T4 = (const float4*)T;
  const float4* W4 = (const float4*)W;

  float sI = 0.0f, sP = 0.0f, sT = 0.0f, sV = 0.0f;
  for (long long i = tid; i < N4; i += stride) {
    float4 p = P4[i], t = T4[i], w = W4[i];
    sI += (p.x * t.x + p.y * t.y) + (p.z * t.z + p.w * t.w);
    sP += (p.x + p.y) + (p.z + p.w);
    sT += (t.x + t.y) + (t.z + t.w);
    sV += ((w.x > 0.0f) ? 1.0f : 0.0f) + ((w.y > 0.0f) ? 1.0f : 0.0f) +
          ((w.z > 0.0f) ? 1.0f : 0.0f) + ((w.w > 0.0f) ? 1.0f : 0.0f);
  }
  const long long base = N4 << 2;  // tail (N % 4 elements)
  if (tid < (N - base)) {
    float p = P[base + tid], t = T[base + tid], w = W[base + tid];
    sI += p * t; sP += p; sT += t; sV += (w > 0.0f) ? 1.0f : 0.0f;
  }

  sI = wave_sum_f32(sI); sP = wave_sum_f32(sP);
  sT = wave_sum_f32(sT); sV = wave_sum_f32(sV);

  __shared__ float red[8][4];
  const int lane = threadIdx.x & 31, wv = threadIdx.x >> 5;
  if (lane == 0) { red[wv][0] = sI; red[wv][1] = sP; red[wv][2] = sT; red[wv][3] = sV; }
  __syncthreads();
  if (threadIdx.x == 0) {
    float4 o; o.x = 0.0f; o.y = 0.0f; o.z = 0.0f; o.w = 0.0f;
    for (int k = 0; k < 8; ++k) {
      o.x += red[k][0]; o.y += red[k][1]; o.z += red[k][2]; o.w += red[k][3];
    }
    part[blockIdx.x] = o;
  }
}

// --------------------------- finalize pass 1 -------------------------------
__global__ __launch_bounds__(256) void ghm_fin1(
    const float4* __restrict__ part, int nPart, float* __restrict__ hdr) {
  float sI = 0.0f, sP = 0.0f, sT = 0.0f, sV = 0.0f;
  for (int i = threadIdx.x; i < nPart; i += 256) {
    float4 v = part[i];
    sI += v.x; sP += v.y; sT += v.z; sV += v.w;
  }
  sI = wave_sum_f32(sI); sP = wave_sum_f32(sP);
  sT = wave_sum_f32(sT); sV = wave_sum_f32(sV);

  __shared__ float red[8][4];
  const int lane = threadIdx.x & 31, wv = threadIdx.x >> 5;
  if (lane == 0) { red[wv][0] = sI; red[wv][1] = sP; red[wv][2] = sT; red[wv][3] = sV; }
  __syncthreads();
  if (threadIdx.x == 0) {
    float I = 0.0f, Sp = 0.0f, St = 0.0f, V = 0.0f;
    for (int k = 0; k < 8; ++k) { I += red[k][0]; Sp += red[k][1]; St += red[k][2]; V += red[k][3]; }
    float S = Sp + St;
    hdr[0] = I;
    hdr[1] = S;
    hdr[2] = fmaxf(V, 1.0f);   // tot
    hdr[3] = 2.0f * I / S;     // scale for g
  }
}

// ------------------------------- pass 2 ------------------------------------
__global__ __launch_bounds__(256) void ghm_pass2(
    const float* __restrict__ P, const float* __restrict__ T,
    const float* __restrict__ W, long long N, const float* __restrict__ hdr,
    unsigned int* __restrict__ gcnt, float* __restrict__ gsum) {
  __shared__ unsigned int hc[8][10];  // per-wave histograms: counts
  __shared__ float        hs[8][10];  // per-wave histograms: sum(p*t)
  for (int i = threadIdx.x; i < 80; i += 256) {
    (&hc[0][0])[i] = 0u;
    (&hs[0][0])[i] = 0.0f;
  }
  __syncthreads();

  const float scale = hdr[3];
  const int wv = threadIdx.x >> 5;
  const long long tid    = (long long)blockIdx.x * 256 + threadIdx.x;
  const long long stride = (long long)gridDim.x * 256;
  const long long N4     = N >> 2;
  const float4* P4 = (const float4*)P;
  const float4* T4 = (const float4*)T;
  const float4* W4 = (const float4*)W;

  for (long long i = tid; i < N4; i += stride) {
    float4 p = P4[i], t = T4[i], w = W4[i];
    float pe[4] = {p.x, p.y, p.z, p.w};
    float te[4] = {t.x, t.y, t.z, t.w};
    float we[4] = {w.x, w.y, w.z, w.w};
#pragma unroll
    for (int k = 0; k < 4; ++k) {
      float g = fabsf(scale * pe[k] - te[k]);
      if ((we[k] > 0.0f) && (g < 1.000001f)) {
        int bi = (int)(g * 10.0f);
        if (bi > 9) bi = 9;
        atomicAdd(&hc[wv][bi], 1u);
        float pt = pe[k] * te[k];
        if (pt != 0.0f) atomicAdd(&hs[wv][bi], pt);
      }
    }
  }
  const long long base = N4 << 2;  // tail
  if (tid < (N - base)) {
    float p = P[base + tid], t = T[base + tid], w = W[base + tid];
    float g = fabsf(scale * p - t);
    if ((w > 0.0f) && (g < 1.000001f)) {
      int bi = (int)(g * 10.0f);
      if (bi > 9) bi = 9;
      atomicAdd(&hc[wv][bi], 1u);
      float pt = p * t;
      if (pt != 0.0f) atomicAdd(&hs[wv][bi], pt);
    }
  }
  __syncthreads();

  if (threadIdx.x < 10) {
    unsigned int c = 0; float s = 0.0f;
    for (int k = 0; k < 8; ++k) { c += hc[k][threadIdx.x]; s += hs[k][threadIdx.x]; }
    gcnt[(size_t)blockIdx.x * 10 + threadIdx.x] = c;
    gsum[(size_t)blockIdx.x * 10 + threadIdx.x] = s;
  }
}

// --------------------------- finalize pass 2 -------------------------------
__global__ __launch_bounds__(320) void ghm_fin2(
    const unsigned int* __restrict__ gcnt, const float* __restrict__ gsum,
    int nPart, const float* __restrict__ hdr, float* __restrict__ out) {
  __shared__ unsigned int ct[10];
  __shared__ float        st[10];
  const int wv = threadIdx.x >> 5, lane = threadIdx.x & 31;
  if (wv < 10) {  // wave-uniform branch: EXEC stays all-ones for WMMA
    unsigned int c = 0; float s = 0.0f;
    for (int i = lane; i < nPart; i += 32) {
      c += gcnt[(size_t)i * 10 + wv];
      s += gsum[(size_t)i * 10 + wv];
    }
    c = wave_sum_u32(c);
    s = wave_sum_f32(s);
    if (lane == 0) { ct[wv] = c; st[wv] = s; }
  }
  __syncthreads();
  if (threadIdx.x == 0) {
    const float S = hdr[1], tot = hdr[2];
    float n = 0.0f;
    for (int b = 0; b < 10; ++b) if (ct[b] > 0u) n += 1.0f;
    const float nm = fmaxf(n, 1.0f);
    float num = 0.0f;
    for (int b = 0; b < 10; ++b) {
      float cb = fmaxf((float)ct[b], 1.0f);
      num += 2.0f * st[b] * ((tot / cb) / nm);
    }
    out[0] = 1.0f - num / S;  // LOSS_WEIGHT = 1.0
  }
}

// ---------------------------------------------------------------------------
extern "C" void kernel_launch(void* const* d_in, const int* in_sizes, int n_in,
                              void* d_out, int out_size, void* d_ws, size_t ws_size,
                              hipStream_t stream) {
  const float* P = (const float*)d_in[0];
  const float* T = (const float*)d_in[1];
  const float* W = (const float*)d_in[2];
  const long long N = (long long)in_sizes[0];

  int grid = 2048;  // ~16 float4 per thread at N = 33.6M; fills the device
  while (grid > 1 && (256 + (size_t)grid * 96) > ws_size) grid >>= 1;

  char* ws = (char*)d_ws;
  float*        hdr  = (float*)ws;                                         // 4 scalars
  float4*       part = (float4*)(ws + 256);                                // grid * 16 B
  unsigned int* gcnt = (unsigned int*)(ws + 256 + (size_t)grid * 16);      // grid * 40 B
  float*        gsum = (float*)(ws + 256 + (size_t)grid * 16 + (size_t)grid * 40);

  ghm_pass1<<<grid, 256, 0, stream>>>(P, T, W, N, part);
  ghm_fin1 <<<1,    256, 0, stream>>>(part, grid, hdr);
  ghm_pass2<<<grid, 256, 0, stream>>>(P, T, W, N, hdr, gcnt, gsum);
  ghm_fin2 <<<1,    320, 0, stream>>>(gcnt, gsum, grid, hdr, (float*)d_out);
}